// DeepMamba_72713796321528
// MI455X (gfx1250) — compile-verified
//
#include <hip/hip_runtime.h>
#include <hip/hip_bf16.h>

#define BB 4
#define LL 256
#define IN_DIM 24
#define DD 1024
#define NLAYERS 4
#define NN 32
#define KK 4
#define DI_ 2048
#define DTR_ 64
#define XPROJ_OUT 128  // DTR + 2N

typedef __attribute__((ext_vector_type(2))) float v2f;
typedef __attribute__((ext_vector_type(8))) float v8f;

__device__ __forceinline__ float sigmoidf_(float x) { return 1.0f / (1.0f + __expf(-x)); }
__device__ __forceinline__ float siluf_(float x) { return x * sigmoidf_(x); }
__device__ __forceinline__ float softplusf_(float x) { return (x > 20.0f) ? x : log1pf(__expf(x)); }

// -----------------------------------------------------------------------------
// fp32 WMMA GEMM:  C[M x N] = X[M x K] (row stride lda) * W[N x K]^T
// Register-blocked: one wave owns a 64x64 output tile (4x4 grid of 16x16 WMMA
// tiles, 128 VGPRs of accumulator). Per K=4 step: 4 A-frags + 4 B-frags feed
// 16 v_wmma_f32_16x16x4_f32 -> 16 FLOP/byte from L2 and 16 independent
// accumulation chains for latency hiding.
// MODE 0: plain store.  MODE 1: bias + softplus epilogue (dt projection).
// Requires: M % 64 == 0, N % 64 == 0, K % 4 == 0 (true for all calls).
// -----------------------------------------------------------------------------
template <int MODE>
__global__ __launch_bounds__(256) void wmma_gemm_f32_kernel(
    const float* __restrict__ X, const float* __restrict__ W,
    const float* __restrict__ bias, float* __restrict__ Cout,
    int N, int K, int lda) {
  const int MT = 4, NT = 4;
  const int lane = threadIdx.x & 31;
  const int wave = (blockIdx.x * blockDim.x + threadIdx.x) >> 5;
  const int tilesN = N / (16 * NT);
  const int tm = wave / tilesN;
  const int tn = wave % tilesN;
  const int lm = lane & 15;
  const int half = lane >> 4;

  v8f acc[MT][NT] = {};

  // A fragment: lanes 0-15 hold K=k,k+1 ; lanes 16-31 hold K=k+2,k+3 (row = M)
  // B fragment: same K split, lane index = N column (B[k][n] = W[n][k])
  const float* __restrict__ xa[MT];
  const float* __restrict__ wb[NT];
#pragma unroll
  for (int mi = 0; mi < MT; ++mi)
    xa[mi] = X + (size_t)(tm * 64 + mi * 16 + lm) * lda + 2 * half;
#pragma unroll
  for (int ni = 0; ni < NT; ++ni) {
    wb[ni] = W + (size_t)(tn * 64 + ni * 16 + lm) * K + 2 * half;
    __builtin_prefetch(wb[ni], 0, 3);  // global_prefetch_b8 warm-up
  }

#pragma unroll 1
  for (int k = 0; k < K; k += 4) {
    v2f a[MT], b[NT];
#pragma unroll
    for (int mi = 0; mi < MT; ++mi) a[mi] = *(const v2f*)(xa[mi] + k);
#pragma unroll
    for (int ni = 0; ni < NT; ++ni) b[ni] = *(const v2f*)(wb[ni] + k);
#pragma unroll
    for (int mi = 0; mi < MT; ++mi)
#pragma unroll
      for (int ni = 0; ni < NT; ++ni)
        acc[mi][ni] = __builtin_amdgcn_wmma_f32_16x16x4_f32(
            false, a[mi], false, b[ni], (short)0, acc[mi][ni], false, false);
  }

#pragma unroll
  for (int ni = 0; ni < NT; ++ni) {
    const int n = tn * 64 + ni * 16 + lm;
    float bv = 0.0f;
    if (MODE == 1) bv = bias[n];
#pragma unroll
    for (int mi = 0; mi < MT; ++mi) {
#pragma unroll
      for (int j = 0; j < 8; ++j) {
        const int r = tm * 64 + mi * 16 + half * 8 + j;  // C/D: VGPR j -> M=j (+8 hi half)
        float v = acc[mi][ni][j];
        if (MODE == 1) v = softplusf_(v + bv);
        Cout[(size_t)r * N + n] = v;
      }
    }
  }
}

// -----------------------------------------------------------------------------
// Depthwise causal conv (K=4, left pad 3) over xi = xz[..., :DI], + bias, SiLU.
// -----------------------------------------------------------------------------
__global__ __launch_bounds__(256) void conv_silu_kernel(
    const float* __restrict__ xz, const float* __restrict__ cw,
    const float* __restrict__ cb, float* __restrict__ xc) {
  const size_t idx = (size_t)blockIdx.x * blockDim.x + threadIdx.x;  // B*L*DI
  const int e = (int)(idx % DI_);
  const size_t bl = idx / DI_;
  const int l = (int)(bl % LL);
  const int b = (int)(bl / LL);
  const float* w = cw + (size_t)e * KK;
  float s = cb[e];
  const float* base = xz + (size_t)b * LL * (2 * DI_) + e;
#pragma unroll
  for (int j = 0; j < KK; ++j) {
    const int lj = l + j - (KK - 1);
    if (lj >= 0) s += w[j] * base[(size_t)lj * (2 * DI_)];
  }
  xc[idx] = siluf_(s);
}

// -----------------------------------------------------------------------------
// Selective scan: one thread per (b, d), 32-element state in registers.
// B/C tiles for this batch staged in 64KB dynamic LDS (shared by all d).
// Epilogue fused: y += x*D ; y *= silu(z) ; store.
// -----------------------------------------------------------------------------
__global__ __launch_bounds__(256) void scan_kernel(
    const float* __restrict__ xc, const float* __restrict__ dt,
    const float* __restrict__ dbc, const float* __restrict__ xz,
    const float* __restrict__ A_log, const float* __restrict__ Dp,
    float* __restrict__ y) {
  extern __shared__ float smem[];  // 2 * L * N floats = 64KB
  float* Bs = smem;
  float* Cs = smem + LL * NN;
  const int tid = threadIdx.x;
  const int chunks = DI_ / 256;
  const int b = blockIdx.x / chunks;
  const int d = (blockIdx.x % chunks) * 256 + tid;

  for (int i = tid; i < LL * NN; i += 256) {
    const int t = i >> 5;
    const int n = i & 31;
    const float* r = dbc + ((size_t)b * LL + t) * XPROJ_OUT + DTR_;
    Bs[i] = r[n];
    Cs[i] = r[NN + n];
  }
  __syncthreads();

  float A[NN];
#pragma unroll
  for (int n = 0; n < NN; ++n) A[n] = -__expf(A_log[(size_t)d * NN + n]);
  float h[NN];
#pragma unroll
  for (int n = 0; n < NN; ++n) h[n] = 0.0f;
  const float dpar = Dp[d];

  for (int t = 0; t < LL; ++t) {
    const size_t off = (size_t)b * LL + t;
    const float dtt = dt[off * DI_ + d];
    const float xt = xc[off * DI_ + d];
    const float dbx = dtt * xt;
    const float* bs = Bs + t * NN;
    const float* cs = Cs + t * NN;
    float yv = 0.0f;
#pragma unroll
    for (int n = 0; n < NN; ++n) {
      const float dA = __expf(dtt * A[n]);
      const float hn = h[n] * dA + dbx * bs[n];
      h[n] = hn;
      yv += hn * cs[n];
    }
    yv += xt * dpar;
    const float z = xz[off * (2 * DI_) + DI_ + d];
    yv *= siluf_(z);
    y[off * DI_ + d] = yv;
  }
}

// -----------------------------------------------------------------------------
// input projection: h = x @ W^T + b   (K = 24, LDS-staged input row)
// grid: (B*L) * 4 blocks; block covers 256 output columns of one row.
// -----------------------------------------------------------------------------
__global__ __launch_bounds__(256) void input_proj_kernel(
    const float* __restrict__ x, const float* __restrict__ W,
    const float* __restrict__ bias, float* __restrict__ h) {
  __shared__ float xs[IN_DIM];
  const int row = blockIdx.x >> 2;
  const int col = ((blockIdx.x & 3) << 8) + threadIdx.x;
  if (threadIdx.x < IN_DIM) xs[threadIdx.x] = x[(size_t)row * IN_DIM + threadIdx.x];
  __syncthreads();
  const float* w = W + (size_t)col * IN_DIM;
  float s = bias[col];
#pragma unroll
  for (int k = 0; k < IN_DIM; ++k) s += xs[k] * w[k];
  h[(size_t)row * DD + col] = s;
}

// -----------------------------------------------------------------------------
// LayerNorm on the last token only (the only rows the output depends on) + head.
// One block per batch element.
// -----------------------------------------------------------------------------
__global__ __launch_bounds__(256) void ln_head_kernel(
    const float* __restrict__ h, const float* __restrict__ g,
    const float* __restrict__ be, const float* __restrict__ hw,
    const float* __restrict__ hb, float* __restrict__ out) {
  __shared__ float r1[256], r2[256];
  const int b = blockIdx.x;
  const int tid = threadIdx.x;
  const float* row = h + ((size_t)b * LL + (LL - 1)) * DD;
  float s = 0.0f, s2 = 0.0f;
  for (int i = tid; i < DD; i += 256) {
    const float v = row[i];
    s += v;
    s2 += v * v;
  }
  r1[tid] = s;
  r2[tid] = s2;
  __syncthreads();
  for (int st = 128; st > 0; st >>= 1) {
    if (tid < st) {
      r1[tid] += r1[tid + st];
      r2[tid] += r2[tid + st];
    }
    __syncthreads();
  }
  const float mu = r1[0] / DD;
  const float var = r2[0] / DD - mu * mu;
  const float inv = rsqrtf(var + 1e-5f);
  float dot = 0.0f;
  for (int i = tid; i < DD; i += 256) {
    const float v = (row[i] - mu) * inv * g[i] + be[i];
    dot += v * hw[i];
  }
  __syncthreads();
  r1[tid] = dot;
  __syncthreads();
  for (int st = 128; st > 0; st >>= 1) {
    if (tid < st) r1[tid] += r1[tid + st];
    __syncthreads();
  }
  if (tid == 0) out[b] = r1[0] + hb[0];
}

// -----------------------------------------------------------------------------
extern "C" void kernel_launch(void* const* d_in, const int* in_sizes, int n_in,
                              void* d_out, int out_size, void* d_ws, size_t ws_size,
                              hipStream_t stream) {
  (void)in_sizes; (void)n_in; (void)out_size; (void)ws_size;
  const float* x    = (const float*)d_in[0];
  const float* ipw  = (const float*)d_in[1];
  const float* ipb  = (const float*)d_in[2];
  const float* inw  = (const float*)d_in[3];
  const float* cw   = (const float*)d_in[4];
  const float* cb   = (const float*)d_in[5];
  const float* xpw  = (const float*)d_in[6];
  const float* dtw  = (const float*)d_in[7];
  const float* dtb  = (const float*)d_in[8];
  const float* alog = (const float*)d_in[9];
  const float* dpar = (const float*)d_in[10];
  const float* opw  = (const float*)d_in[11];
  const float* lng  = (const float*)d_in[12];
  const float* lnb  = (const float*)d_in[13];
  const float* hw   = (const float*)d_in[14];
  const float* hb   = (const float*)d_in[15];
  float* out = (float*)d_out;

  const size_t ROWS = (size_t)BB * LL;  // 1024
  float* ws = (float*)d_ws;
  float* h_buf   = ws; ws += ROWS * DD;         // 1M floats
  float* xz_buf  = ws; ws += ROWS * 2 * DI_;    // 4M floats
  float* xc_buf  = ws; ws += ROWS * DI_;        // 2M floats
  float* dbc_buf = ws; ws += ROWS * XPROJ_OUT;  // 128K floats
  float* dt_buf  = ws; ws += ROWS * DI_;        // 2M floats
  float* y_buf   = ws; ws += ROWS * DI_;        // 2M floats

  input_proj_kernel<<<(int)(ROWS * 4), 256, 0, stream>>>(x, ipw, ipb, h_buf);

  for (int i = 0; i < NLAYERS; ++i) {
    const float* inw_i = inw + (size_t)i * 2 * DI_ * DD;
    const float* cw_i  = cw  + (size_t)i * DI_ * KK;
    const float* cb_i  = cb  + (size_t)i * DI_;
    const float* xpw_i = xpw + (size_t)i * XPROJ_OUT * DI_;
    const float* dtw_i = dtw + (size_t)i * DI_ * DTR_;
    const float* dtb_i = dtb + (size_t)i * DI_;
    const float* al_i  = alog + (size_t)i * DI_ * NN;
    const float* dp_i  = dpar + (size_t)i * DI_;
    const float* opw_i = opw + (size_t)i * DD * DI_;

    // GEMM1: xz = h @ inw^T   (M=1024, N=4096, K=1024) -> 1024 waves
    {
      const int Nn = 2 * DI_, Kk = DD;
      const int waves = (int)(ROWS / 64) * (Nn / 64);
      wmma_gemm_f32_kernel<0><<<waves * 32 / 256, 256, 0, stream>>>(
          h_buf, inw_i, nullptr, xz_buf, Nn, Kk, DD);
    }
    // depthwise conv + bias + SiLU
    conv_silu_kernel<<<(int)((ROWS * DI_) / 256), 256, 0, stream>>>(
        xz_buf, cw_i, cb_i, xc_buf);
    // GEMM2: dbc = xc @ xpw^T  (N=128, K=2048) -> 32 waves
    {
      const int Nn = XPROJ_OUT, Kk = DI_;
      const int waves = (int)(ROWS / 64) * (Nn / 64);
      wmma_gemm_f32_kernel<0><<<waves * 32 / 256, 256, 0, stream>>>(
          xc_buf, xpw_i, nullptr, dbc_buf, Nn, Kk, DI_);
    }
    // GEMM3: dt = softplus(dbc[:, :64] @ dtw^T + dtb)  (N=2048, K=64, lda=128)
    {
      const int Nn = DI_, Kk = DTR_;
      const int waves = (int)(ROWS / 64) * (Nn / 64);
      wmma_gemm_f32_kernel<1><<<waves * 32 / 256, 256, 0, stream>>>(
          dbc_buf, dtw_i, dtb_i, dt_buf, Nn, Kk, XPROJ_OUT);
    }
    // selective scan (fused D-skip and SiLU(z) gate)
    scan_kernel<<<BB * (DI_ / 256), 256, 2 * LL * NN * sizeof(float), stream>>>(
        xc_buf, dt_buf, dbc_buf, xz_buf, al_i, dp_i, y_buf);
    // GEMM4: h = y @ opw^T  (N=1024, K=2048) -> 256 waves
    {
      const int Nn = DD, Kk = DI_;
      const int waves = (int)(ROWS / 64) * (Nn / 64);
      wmma_gemm_f32_kernel<0><<<waves * 32 / 256, 256, 0, stream>>>(
          y_buf, opw_i, nullptr, h_buf, Nn, Kk, DI_);
    }
  }

  ln_head_kernel<<<BB, 256, 0, stream>>>(h_buf, lng, lnb, hw, hb, out);
}